// NumProj_20598663151715
// MI455X (gfx1250) — compile-verified
//
#include <hip/hip_runtime.h>

typedef __attribute__((ext_vector_type(2))) float v2f;
typedef __attribute__((ext_vector_type(8))) float v8f;

#define N_NUM 200
#define EMB   64
#define BATCH 16384

// out[b,i,e] = x[b,i] * W[i,e] + bias[i,e]
// One wave handles a 16-batch tile for one feature i, covering all 64 emb cols
// as 4 WMMA 16x16 tiles (V_WMMA_F32_16X16X4_F32, rank-1 + bias form).
__global__ __launch_bounds__(256) void numproj_wmma_kernel(
    const float* __restrict__ x,
    const float* __restrict__ W,
    const float* __restrict__ bias,
    float* __restrict__ out)
{
    const int lane  = threadIdx.x & 31;
    const int waveN = threadIdx.x >> 5;
    const int w     = blockIdx.x * 8 + waveN;      // global wave id
    const int i     = w % N_NUM;                   // feature index
    const int btile = w / N_NUM;                   // batch tile index
    const int b0    = btile << 4;

    const int  ml = lane & 15;
    const bool lo = lane < 16;

    // ---- A matrix (16x4 f32): only K=0 column nonzero = x[b0+m, i].
    // Layout: A-VGPR0 lanes0-15 hold K=0 (M=lane). Load full-EXEC, mask via cndmask.
    const float xval = x[(size_t)(b0 + ml) * N_NUM + i];
    v2f a;
    a.x = lo ? xval : 0.0f;   // K=0 (lanes 0-15); lanes 16-31 hold K=2 -> zero
    a.y = 0.0f;               // K=1 / K=3 -> zero

    const float* Wrow = W    + (size_t)i * EMB;
    const float* Brow = bias + (size_t)i * EMB;

    const size_t rowStride = (size_t)N_NUM * EMB;  // 12800 floats between batch rows
    const int    mrow      = (lane >> 4) << 3;     // 0 or 8 (C/D layout M offset)
    float* orow = out + (size_t)b0 * rowStride + (size_t)i * EMB;

#pragma unroll
    for (int t = 0; t < 4; ++t) {
        const int e0 = t << 4;

        // ---- B matrix (4x16 f32): only K=0 row nonzero = W[i, e0+n].
        const float wval = Wrow[e0 + ml];
        v2f bm;
        bm.x = lo ? wval : 0.0f;
        bm.y = 0.0f;

        // ---- C matrix: bias broadcast down M. For every C VGPR, N = lane%16,
        // so each lane holds the same bias value in all 8 accumulator regs.
        const float bv = Brow[e0 + ml];
        v8f c;
#pragma unroll
        for (int k = 0; k < 8; ++k) c[k] = bv;

        // D = A x B + C  -> D[m,n] = x[b0+m,i]*W[i,e0+n] + bias[i,e0+n]
        v8f d = __builtin_amdgcn_wmma_f32_16x16x4_f32(
            /*neg_a=*/false, a, /*neg_b=*/false, bm,
            /*c_mod=*/(short)0, c, /*reuse_a=*/false, /*reuse_b=*/false);

        // ---- Store per C/D layout: VGPR k, lane l -> (M = k + 8*(l/16), N = l%16).
        // Each store = two contiguous 64B segments; output never re-read -> NT.
        float* op = orow + (size_t)mrow * rowStride + e0 + ml;
#pragma unroll
        for (int k = 0; k < 8; ++k) {
            __builtin_nontemporal_store(d[k], op + (size_t)k * rowStride);
        }
    }
}

extern "C" void kernel_launch(void* const* d_in, const int* in_sizes, int n_in,
                              void* d_out, int out_size, void* d_ws, size_t ws_size,
                              hipStream_t stream) {
    (void)in_sizes; (void)n_in; (void)out_size; (void)d_ws; (void)ws_size;
    const float* x    = (const float*)d_in[0];
    const float* W    = (const float*)d_in[1];
    const float* bias = (const float*)d_in[2];
    float* out = (float*)d_out;

    // (16384/16) b-tiles * 200 features = 204800 waves; 8 waves (256 thr) / block.
    const int blocks = (BATCH / 16) * N_NUM / 8;   // 25600, exact -> no tail, EXEC all-1s
    numproj_wmma_kernel<<<blocks, 256, 0, stream>>>(x, W, bias, out);
}